// PairwiseSimilarityLoss_31748398252479
// MI455X (gfx1250) — compile-verified
//
#include <hip/hip_runtime.h>

// CDNA5 (gfx1250) Matryoshka pairwise-cosine loss.
// Full-precision path: V_WMMA_F32_16X16X4_F32 (f32 matrix cores, wave32),
// with 2x2 register blocking: each wave owns a 32x32 output super-tile so
// every operand load feeds two WMMAs (1 load per WMMA instead of 2).

typedef __attribute__((ext_vector_type(2))) float v2f;
typedef __attribute__((ext_vector_type(8))) float v8f;

#define NORM_THREADS 256
#define WAVES_PER_BLOCK 8
#define STILE 32  // super-tile edge (2x2 grid of 16x16 WMMA tiles)

__device__ __forceinline__ v8f wmma_f32(v2f a, v2f b, v8f c) {
  return __builtin_amdgcn_wmma_f32_16x16x4_f32(false, a, false, b,
                                               (short)0, c, false, false);
}

// ---------------------------------------------------------------------------
// Kernel 1: reciprocal row norms.
//   invE[i]        = 1/||E[i, 0:1024]||
//   invA[mi*N + i] = 1/||A[i, 0:m]||  for m in {64,128,256,512}
// ---------------------------------------------------------------------------
__device__ __forceinline__ float block_reduce_sum(float v, float* sh, int t) {
  sh[t] = v;
  __syncthreads();
#pragma unroll
  for (int off = NORM_THREADS / 2; off > 0; off >>= 1) {
    if (t < off) sh[t] += sh[t + off];
    __syncthreads();
  }
  float r = sh[0];
  __syncthreads();
  return r;
}

__global__ __launch_bounds__(NORM_THREADS) void mrl_norms_kernel(
    const float* __restrict__ E, const float* __restrict__ A,
    float* __restrict__ invE, float* __restrict__ invA, int N, int d) {
  __shared__ float sh[NORM_THREADS];
  const int row = blockIdx.x;
  const int t   = threadIdx.x;

  const float* e = E + (size_t)row * d;
  float s = 0.f;
  for (int k = t; k < d; k += NORM_THREADS) { float v = e[k]; s += v * v; }
  s = block_reduce_sum(s, sh, t);
  if (t == 0) invE[row] = rsqrtf(s);

  const float* a = A + (size_t)row * d;
  const float v1 = a[t];
  const float v2 = a[t + NORM_THREADS];
  const float q1 = v1 * v1;
  const float q2 = v2 * v2;
  const float s64  = block_reduce_sum((t < 64)  ? q1 : 0.f, sh, t);
  const float s128 = block_reduce_sum((t < 128) ? q1 : 0.f, sh, t);
  const float s256 = block_reduce_sum(q1, sh, t);
  const float s512 = s256 + block_reduce_sum(q2, sh, t);
  if (t == 0) {
    invA[0 * N + row] = rsqrtf(s64);
    invA[1 * N + row] = rsqrtf(s128);
    invA[2 * N + row] = rsqrtf(s256);
    invA[3 * N + row] = rsqrtf(s512);
  }
}

// ---------------------------------------------------------------------------
// Kernel 2: one wave32 per 32x32 super-tile (four 16x16 WMMA sub-tiles).
// E-Gram over K=1024 into accE[2][2]; target cosines cached in registers;
// A-Gram over K=512 with checkpoints at m = 64/128/256/512 accumulating
// weighted |cos_m - target|.  Symmetry: super-tiles sti<=stj only, with
// per-element weight 2 (i<j), 1 (i==j), 0 (i>j or i+j>=N).
//
// Operand layout (ISA 7.12.2, 32-bit A 16x4): lanes 0-15 hold K+0/K+1,
// lanes 16-31 hold K+2/K+3 for the row selected by (lane & 15).
// C/D 16x16 f32: vgpr e, lane l -> row base + 8*(l>=16) + e, col base+(l&15).
// ---------------------------------------------------------------------------
__global__ __launch_bounds__(WAVES_PER_BLOCK * 32) void mrl_gram_kernel(
    const float* __restrict__ E, const float* __restrict__ A,
    const float* __restrict__ invE, const float* __restrict__ invA,
    float* __restrict__ sums, int N, int d) {
  const int wave = threadIdx.x >> 5;
  const int lane = threadIdx.x & 31;

  const int S = N / STILE;  // 128 super-tiles per side
  const int st = blockIdx.x * WAVES_PER_BLOCK + wave;
  const int sti = st / S;
  const int stj = st - sti * S;
  if (sti > stj) return;        // symmetry: upper triangle of super-tiles
  if (sti + stj >= S) return;   // fully outside mask i+j<N

  const int i0 = sti * STILE;
  const int j0 = stj * STILE;
  const int lo = lane & 15;
  const int hi = lane >> 4;

  const v8f vzero = {};

  // ---- E Gram over K = 0..d, 2x2 blocked (4 independent WMMA chains) ----
  const float* pa0 = E + (size_t)(i0 + lo) * d + 2 * hi;
  const float* pa1 = pa0 + (size_t)16 * d;
  const float* pb0 = E + (size_t)(j0 + lo) * d + 2 * hi;
  const float* pb1 = pb0 + (size_t)16 * d;

  v8f acc[2][2];
  acc[0][0] = vzero; acc[0][1] = vzero; acc[1][0] = vzero; acc[1][1] = vzero;

#pragma unroll 4
  for (int k = 0; k < d; k += 4) {
    v2f a0 = *(const v2f*)(pa0 + k);
    v2f a1 = *(const v2f*)(pa1 + k);
    v2f b0 = *(const v2f*)(pb0 + k);
    v2f b1 = *(const v2f*)(pb1 + k);
    acc[0][0] = wmma_f32(a0, b0, acc[0][0]);
    acc[0][1] = wmma_f32(a0, b1, acc[0][1]);
    acc[1][0] = wmma_f32(a1, b0, acc[1][0]);
    acc[1][1] = wmma_f32(a1, b1, acc[1][1]);
  }

  // ---- target cosines (kept live in registers across the A phase) ----
  const int jc[2] = {j0 + lo, j0 + 16 + lo};
  const float invEj[2] = {invE[jc[0]], invE[jc[1]]};
  float tgt[2][2][8];
#pragma unroll
  for (int ib = 0; ib < 2; ++ib) {
#pragma unroll
    for (int e2 = 0; e2 < 8; ++e2) {
      const int i = i0 + ib * 16 + hi * 8 + e2;
      const float invEi = invE[i];
#pragma unroll
      for (int jb = 0; jb < 2; ++jb)
        tgt[ib][jb][e2] = acc[ib][jb][e2] * invEi * invEj[jb];
    }
  }

  // ---- A Gram over K = 0..512 with Matryoshka checkpoints ----
  pa0 = A + (size_t)(i0 + lo) * d + 2 * hi;
  pa1 = pa0 + (size_t)16 * d;
  pb0 = A + (size_t)(j0 + lo) * d + 2 * hi;
  pb1 = pb0 + (size_t)16 * d;
  acc[0][0] = vzero; acc[0][1] = vzero; acc[1][0] = vzero; acc[1][1] = vzero;

  const int mend[4] = {64, 128, 256, 512};
  float psum[4];
  int kk = 0;
#pragma unroll
  for (int mi = 0; mi < 4; ++mi) {
#pragma unroll 4
    for (; kk < mend[mi]; kk += 4) {
      v2f a0 = *(const v2f*)(pa0 + kk);
      v2f a1 = *(const v2f*)(pa1 + kk);
      v2f b0 = *(const v2f*)(pb0 + kk);
      v2f b1 = *(const v2f*)(pb1 + kk);
      acc[0][0] = wmma_f32(a0, b0, acc[0][0]);
      acc[0][1] = wmma_f32(a0, b1, acc[0][1]);
      acc[1][0] = wmma_f32(a1, b0, acc[1][0]);
      acc[1][1] = wmma_f32(a1, b1, acc[1][1]);
    }
    const float invAj[2] = {invA[mi * N + jc[0]], invA[mi * N + jc[1]]};
    float s = 0.f;
#pragma unroll
    for (int ib = 0; ib < 2; ++ib) {
#pragma unroll
      for (int e2 = 0; e2 < 8; ++e2) {
        const int i = i0 + ib * 16 + hi * 8 + e2;
        const float invAi = invA[mi * N + i];
#pragma unroll
        for (int jb = 0; jb < 2; ++jb) {
          const int j = jc[jb];
          const float m = ((i + j) < N) ? 1.f : 0.f;
          const float w = m * ((i < j) ? 2.f : ((i == j) ? 1.f : 0.f));
          const float c = acc[ib][jb][e2] * invAi * invAj[jb];
          s += w * fabsf(c - tgt[ib][jb][e2]);
        }
      }
    }
    psum[mi] = s;
  }

  // ---- wave32 reduction + global accumulation ----
#pragma unroll
  for (int mi = 0; mi < 4; ++mi) {
    float s = psum[mi];
#pragma unroll
    for (int off = 16; off > 0; off >>= 1) s += __shfl_down(s, off, 32);
    if (lane == 0) atomicAdd(&sums[mi], s);
  }
}

// ---------------------------------------------------------------------------
// Kernel 3: finalize -> out = [loss, partial_loss[0..3]]
// ---------------------------------------------------------------------------
__global__ void mrl_finalize_kernel(const float* __restrict__ sums,
                                    float* __restrict__ out, float invCounter) {
  if (threadIdx.x == 0 && blockIdx.x == 0) {
    const float s0 = sums[0], s1 = sums[1], s2 = sums[2], s3 = sums[3];
    out[0] = (s0 + s1 + s2 + s3) * invCounter;
    out[1] = s0 * invCounter;
    out[2] = s1 * invCounter;
    out[3] = s2 * invCounter;
    out[4] = s3 * invCounter;
  }
}

// ---------------------------------------------------------------------------
extern "C" void kernel_launch(void* const* d_in, const int* in_sizes, int n_in,
                              void* d_out, int out_size, void* d_ws, size_t ws_size,
                              hipStream_t stream) {
  const float* E = (const float*)d_in[0];  // embeddings (N, d) f32
  const float* A = (const float*)d_in[1];  // adapted embeddings (N, d) f32
  const int d = 1024;                      // fixed by reference
  const int N = in_sizes[0] / d;           // 4096

  // Workspace: invE[N] | invA[4*N] | sums[4]
  float* invE = (float*)d_ws;
  float* invA = invE + N;
  float* sums = invA + 4 * N;
  hipMemsetAsync(sums, 0, 4 * sizeof(float), stream);

  mrl_norms_kernel<<<N, NORM_THREADS, 0, stream>>>(E, A, invE, invA, N, d);

  const int S = N / STILE;                 // 128
  const int nSuper = S * S;                // 16384
  const int nBlocks = (nSuper + WAVES_PER_BLOCK - 1) / WAVES_PER_BLOCK;
  mrl_gram_kernel<<<nBlocks, WAVES_PER_BLOCK * 32, 0, stream>>>(
      E, A, invE, invA, sums, N, d);

  // counter = |M_LIST| * N*(N+1)/2 (number of (i,j,m) triples with i+j<N)
  const double counter = (double)N * (double)(N + 1) * 0.5 * 4.0;
  mrl_finalize_kernel<<<1, 32, 0, stream>>>(sums, (float*)d_out,
                                            (float)(1.0 / counter));
}